// Net2DSeg_63479616635457
// MI455X (gfx1250) — compile-verified
//
#include <hip/hip_runtime.h>

// ---------------- problem constants (from reference) ----------------
#define B_     8
#define C_     64        // channels / hidden dim
#define H_     376
#define W_     1241
#define NPTS   30000
#define DIN    80
#define NCLS   19
#define NROWS  (B_ * NPTS)      // 240000

#define WAVES           4
#define ROWS_PER_BLOCK  (WAVES * 16)   // 64 -> 3750 blocks exactly

// Padded LDS row strides (DWORDs). Chosen so that stride mod 64 generates 16
// distinct banks across a 16-lane half-wave (stride residues 20, 4, 4, 4):
#define W1T_STRIDE   84   // >= DIN(80);  84 % 64 = 20, gcd(20,64)=4 -> conflict-free
#define W2T_STRIDE  132   // >= 2C(128); 132 % 64 = 4
#define WCT_STRIDE   68   // >= C(64);    68 % 64 = 4
#define TILE_STRIDE  68   // >= C(64)

typedef __attribute__((ext_vector_type(2))) float v2f;
typedef __attribute__((ext_vector_type(8))) float v8f;

// CDNA5 fp32 WMMA: D[16x16] = A[16x4] * B[4x16] + C   (exact fp32 math)
__device__ __forceinline__ v8f wmma_f32(v2f a, v2f b, v8f c) {
    return __builtin_amdgcn_wmma_f32_16x16x4_f32(
        /*neg_a=*/false, a, /*neg_b=*/false, b,
        /*c_mod=*/(short)0, c, /*reuse_a=*/false, /*reuse_b=*/false);
}

__global__ __launch_bounds__(WAVES * 32)
void net2dseg_fused_kernel(const float* __restrict__ feats_full,   // [B,C,H,W]
                           const float* __restrict__ fusion,       // [NROWS,DIN]
                           const int*   __restrict__ img_idx,      // [B,NPTS,2]
                           const float* __restrict__ W1,           // [DIN,C]
                           const float* __restrict__ b1,           // [C]
                           const float* __restrict__ W2,           // [2C,C]
                           const float* __restrict__ b2,           // [C]
                           const float* __restrict__ Wc1,          // [C,NCLS]
                           const float* __restrict__ bc1,          // [NCLS]
                           const float* __restrict__ Wc2,          // [C,NCLS]
                           const float* __restrict__ bc2,          // [NCLS]
                           float* __restrict__ out1,               // [NROWS,NCLS]
                           float* __restrict__ out2)               // [NROWS,NCLS]
{
    // ---- LDS: weights staged TRANSPOSED ([N][K], padded) + per-wave tiles ----
    __shared__ __align__(16) float sW1t [C_ * W1T_STRIDE];        // 21.0 KB
    __shared__ __align__(16) float sW2t [C_ * W2T_STRIDE];        // 33.0 KB
    __shared__ __align__(16) float sWc1t[32 * WCT_STRIDE];        // 8.5 KB (zero-padded n>=19)
    __shared__ __align__(16) float sWc2t[32 * WCT_STRIDE];        // 8.5 KB
    __shared__ __align__(16) float sFeat[WAVES][16 * TILE_STRIDE]; // gathered feats tiles
    __shared__ __align__(16) float sTile[WAVES][16 * TILE_STRIDE]; // f, then reused for g

    const int tid  = threadIdx.x;
    const int nt   = blockDim.x;
    const int lane = tid & 31;
    const int wave = tid >> 5;

    // cooperative weight staging, transposed so B fragments are contiguous K pairs
    for (int i = tid; i < DIN * C_; i += nt) {
        const int k = i / C_, n = i - k * C_;            // W1[k][n]
        sW1t[n * W1T_STRIDE + k] = W1[i];
    }
    for (int i = tid; i < 2 * C_ * C_; i += nt) {
        const int k = i / C_, n = i - k * C_;            // W2[k][n]
        sW2t[n * W2T_STRIDE + k] = W2[i];
    }
    for (int i = tid; i < 32 * C_; i += nt) {            // zero-pad cls cols 19..31
        const int n = i / C_, k = i - n * C_;
        const float v1 = (n < NCLS) ? Wc1[k * NCLS + n] : 0.0f;
        const float v2 = (n < NCLS) ? Wc2[k * NCLS + n] : 0.0f;
        sWc1t[n * WCT_STRIDE + k] = v1;
        sWc2t[n * WCT_STRIDE + k] = v2;
    }
    __syncthreads();

    const int row0 = blockIdx.x * ROWS_PER_BLOCK + wave * 16;

    // ---- gather feats_2d tile [16][64] (the bandwidth-bound part) ----
    // Per row: 64 strided channel reads (stride H*W floats) -> LDS row-major.
    const size_t HW = (size_t)H_ * W_;
    for (int r = 0; r < 16; ++r) {
        const int grow = row0 + r;
        const int b    = grow / NPTS;
        const int h    = img_idx[(size_t)grow * 2 + 0];
        const int w    = img_idx[(size_t)grow * 2 + 1];
        const float* src = feats_full + (size_t)b * C_ * HW + (size_t)h * W_ + w;
        sFeat[wave][r * TILE_STRIDE + lane]      = src[(size_t)lane * HW];
        sFeat[wave][r * TILE_STRIDE + lane + 32] = src[(size_t)(lane + 32) * HW];
    }

    // lane -> fragment coordinates (ISA 7.12.2 layouts, wave32)
    const int mrow = lane & 15;            // row within 16-row tile / B column id
    const int koff = (lane >> 4) * 2;      // K sub-offset: 0 for lanes 0-15, 2 for 16-31
    const int mhi  = (lane >> 4) * 8;      // C/D row offset

    // ================= GEMM1: f = relu(fusion @ W1 + b1) =================
    v8f accF[4] = {v8f{}, v8f{}, v8f{}, v8f{}};
    const float* arow = fusion + (size_t)(row0 + mrow) * DIN;
    for (int k0 = 0; k0 < DIN; k0 += 4) {
        v2f a = *(const v2f*)(arow + k0 + koff);               // A[16x4] fragment
        #pragma unroll
        for (int n = 0; n < 4; ++n) {
            const int col = n * 16 + mrow;
            v2f bf = *(const v2f*)&sW1t[col * W1T_STRIDE + k0 + koff]; // one ds_load_b64
            accF[n] = wmma_f32(a, bf, accF[n]);
        }
    }
    // bias + relu, spill f tile to LDS (row-major, padded stride)
    #pragma unroll
    for (int n = 0; n < 4; ++n) {
        const int col = n * 16 + mrow;
        const float bias = b1[col];
        #pragma unroll
        for (int r = 0; r < 8; ++r) {
            float v = accF[n][r] + bias;
            v = v > 0.0f ? v : 0.0f;
            sTile[wave][(r + mhi) * TILE_STRIDE + col] = v;
        }
    }

    // ====== GEMM2: g = feats_2d @ W2[0:64] + f @ W2[64:128] + b2 ======
    v8f accG[4] = {v8f{}, v8f{}, v8f{}, v8f{}};
    for (int k0 = 0; k0 < C_; k0 += 4) {
        v2f a1 = *(const v2f*)&sFeat[wave][mrow * TILE_STRIDE + k0 + koff];
        v2f a2 = *(const v2f*)&sTile[wave][mrow * TILE_STRIDE + k0 + koff];
        #pragma unroll
        for (int n = 0; n < 4; ++n) {
            const int col = n * 16 + mrow;
            v2f bfa = *(const v2f*)&sW2t[col * W2T_STRIDE + k0 + koff];
            v2f bfb = *(const v2f*)&sW2t[col * W2T_STRIDE + C_ + k0 + koff];
            accG[n] = wmma_f32(a1, bfa, accG[n]);
            accG[n] = wmma_f32(a2, bfb, accG[n]);
        }
    }
    // bias, overwrite sTile with g (in-wave DS ordering makes this safe)
    #pragma unroll
    for (int n = 0; n < 4; ++n) {
        const int col = n * 16 + mrow;
        const float bias = b2[col];
        #pragma unroll
        for (int r = 0; r < 8; ++r)
            sTile[wave][(r + mhi) * TILE_STRIDE + col] = accG[n][r] + bias;
    }

    // ===== heads: out1 = g @ Wc1 + bc1 ; out2 = feats_2d @ Wc2 + bc2 =====
    v8f accH1[2] = {v8f{}, v8f{}};
    v8f accH2[2] = {v8f{}, v8f{}};
    for (int k0 = 0; k0 < C_; k0 += 4) {
        v2f ag = *(const v2f*)&sTile[wave][mrow * TILE_STRIDE + k0 + koff];
        v2f af = *(const v2f*)&sFeat[wave][mrow * TILE_STRIDE + k0 + koff];
        #pragma unroll
        for (int n = 0; n < 2; ++n) {
            const int col = n * 16 + mrow;   // zero-padded weights: no branch needed
            v2f bf1 = *(const v2f*)&sWc1t[col * WCT_STRIDE + k0 + koff];
            v2f bf2 = *(const v2f*)&sWc2t[col * WCT_STRIDE + k0 + koff];
            accH1[n] = wmma_f32(ag, bf1, accH1[n]);
            accH2[n] = wmma_f32(af, bf2, accH2[n]);
        }
    }

    // epilogue: bias + store both logit tensors (divergence only here)
    #pragma unroll
    for (int n = 0; n < 2; ++n) {
        const int col = n * 16 + mrow;
        if (col < NCLS) {
            const float bb1 = bc1[col];
            const float bb2 = bc2[col];
            #pragma unroll
            for (int r = 0; r < 8; ++r) {
                const size_t o = (size_t)(row0 + r + mhi) * NCLS + col;
                out1[o] = accH1[n][r] + bb1;
                out2[o] = accH2[n][r] + bb2;
            }
        }
    }
}

extern "C" void kernel_launch(void* const* d_in, const int* in_sizes, int n_in,
                              void* d_out, int out_size, void* d_ws, size_t ws_size,
                              hipStream_t stream) {
    const float* feats_full = (const float*)d_in[0];
    const float* fusion     = (const float*)d_in[1];
    const int*   img_idx    = (const int*)  d_in[2];
    const float* W1  = (const float*)d_in[3];
    const float* b1  = (const float*)d_in[4];
    const float* W2  = (const float*)d_in[5];
    const float* b2  = (const float*)d_in[6];
    const float* Wc1 = (const float*)d_in[7];
    const float* bc1 = (const float*)d_in[8];
    const float* Wc2 = (const float*)d_in[9];
    const float* bc2 = (const float*)d_in[10];

    float* out1 = (float*)d_out;                      // seg_logit  [NROWS, NCLS]
    float* out2 = out1 + (size_t)NROWS * NCLS;        // seg_logit2 [NROWS, NCLS]

    dim3 grid(NROWS / ROWS_PER_BLOCK);                // 3750
    dim3 block(WAVES * 32);                           // 128 threads = 4 waves

    hipLaunchKernelGGL(net2dseg_fused_kernel, grid, block, 0, stream,
                       feats_full, fusion, img_idx,
                       W1, b1, W2, b2, Wc1, bc1, Wc2, bc2,
                       out1, out2);
}